// HWlayer2D_45346264711532
// MI455X (gfx1250) — compile-verified
//
#include <hip/hip_runtime.h>

// Problem constants (from reference): B=16, C=64, H=128, W=128, K=16
#define BN    16
#define CN    64
#define HN    128
#define WN    128
#define KN    16
#define PLANE (HN * WN)                 // 16384 floats per (b,c) plane
#define TPB   256                       // 8 waves (wave32)
#define FLOATS_PER_ITER (TPB * 4)       // 1024 floats per iteration
#define ITERS (PLANE / FLOATS_PER_ITER) // 16
#define STAGE_BYTES (TPB * 16)          // 4096 B per LDS stage

typedef float v4f __attribute__((ext_vector_type(4)));

__device__ __forceinline__ float bperm_f(int byteAddr, float v) {
    // wave32 lane gather on the permute network (no LDS memory touched)
    return __int_as_float(__builtin_amdgcn_ds_bpermute(byteAddr, __float_as_int(v)));
}

// Branchless binary search over 15 sorted midpoints held in lanes 0..14 of
// mlane, then gather focus from lanes 0..15 of flane. 'a' tracks idx*4 (the
// byte address for ds_bpermute). 4 compares total instead of 15; the backend
// interleaves the four independent per-element chains by itself.
__device__ __forceinline__ float nearest_focus(float x, float mlane, float flane,
                                               float m7) {
    int a = (x > m7) ? 32 : 0;            // step 8: compare m[7]
    float mv = bperm_f(a + 12, mlane);    // step 4: compare m[idx+3]
    a += (x > mv) ? 16 : 0;
    mv = bperm_f(a + 4, mlane);           // step 2: compare m[idx+1]
    a += (x > mv) ? 8 : 0;
    mv = bperm_f(a, mlane);               // step 1: compare m[idx]
    a += (x > mv) ? 4 : 0;
    return bperm_f(a, flane);             // focus[idx], idx in [0,15]
}

__global__ __launch_bounds__(TPB) void hwlayer2d_kernel(
    const float* __restrict__ x,
    const float* __restrict__ evaluate,
    const float* __restrict__ focus,
    float* __restrict__ out)
{
    __shared__ v4f lbuf[2][TPB];          // 8 KB: two async-load stages

    const int tid   = threadIdx.x;
    const int plane = blockIdx.x;         // (B,C,H,W) => c = plane % C
    const int c     = plane % CN;
    const int l16   = tid & 15;

    // ---- lane-resident per-channel tables (one tiny vector load each) ----
    const float* evRow = evaluate + c * KN;
    float eL = evRow[l16];
    float eH = evRow[l16 < (KN - 1) ? l16 + 1 : (KN - 1)];
    float mlane = 0.5f * (eL + eH);           // lane k (mod 16) holds midpoint m[k]
    float flane = focus[c * KN + l16];        // lane k (mod 16) holds focus[c][k]
    float m7    = bperm_f(7 << 2, mlane);     // root of the search tree (uniform)

    // ---- async global -> LDS streaming setup ----
    const unsigned long long planeBaseB =
        (unsigned long long)(size_t)x + (unsigned long long)plane * PLANE * 4ull;
    const unsigned ldsBase = (unsigned)(size_t)(&lbuf[0][0]); // LDS byte offset
    const unsigned vOff    = (unsigned)tid * 16u;             // per-lane 16B slice

    {   // prologue: stage 0 of iteration 0
        unsigned lds0 = ldsBase + vOff;
        unsigned long long g0 = planeBaseB;
        asm volatile("global_load_async_to_lds_b128 %0, %1, %2"
                     :: "v"(lds0), "v"(vOff), "s"(g0) : "memory");
    }

    float* outPlane = out + (size_t)plane * PLANE;

#pragma unroll
    for (int i = 0; i < ITERS; ++i) {
        // prefetch next tile into the other stage, then wait for current tile
        if (i + 1 < ITERS) {
            unsigned ldsN = ldsBase + (unsigned)(((i + 1) & 1) * STAGE_BYTES) + vOff;
            unsigned long long gN =
                planeBaseB + (unsigned long long)(i + 1) * (FLOATS_PER_ITER * 4ull);
            asm volatile("global_load_async_to_lds_b128 %0, %1, %2"
                         :: "v"(ldsN), "v"(vOff), "s"(gN) : "memory");
            asm volatile("s_wait_asynccnt 0x1" ::: "memory"); // oldest (current) done
        } else {
            asm volatile("s_wait_asynccnt 0x0" ::: "memory");
        }

        // each wave reads back only the slice it loaded: ASYNCcnt is per-wave
        // and async loads complete in order -> no barrier needed.
        v4f xv = lbuf[i & 1][tid];            // ds_load_b128

        v4f r;
#pragma unroll
        for (int l = 0; l < 4; ++l)
            r[l] = nearest_focus(xv[l], mlane, flane, m7);

        // streamed single-use output: non-temporal B128 store
        __builtin_nontemporal_store(r, (v4f*)(outPlane + i * FLOATS_PER_ITER + tid * 4));
    }
}

extern "C" void kernel_launch(void* const* d_in, const int* in_sizes, int n_in,
                              void* d_out, int out_size, void* d_ws, size_t ws_size,
                              hipStream_t stream)
{
    const float* x        = (const float*)d_in[0];
    const float* evaluate = (const float*)d_in[1];
    const float* focus    = (const float*)d_in[2];
    float*       out      = (float*)d_out;

    dim3 grid(BN * CN);   // 1024 blocks, one (b,c) plane each
    dim3 block(TPB);      // 256 threads = 8 wave32 waves
    hwlayer2d_kernel<<<grid, block, 0, stream>>>(x, evaluate, focus, out);
}